// HardDTW_43739946943557
// MI455X (gfx1250) — compile-verified
//
#include <hip/hip_runtime.h>
#include <cstdint>
#include <cstddef>

// Problem constants (fixed by the reference).
#define B_       32
#define T_       2048
#define K_       1024
#define NTHREADS 256            // 8 wave32s per block
#define CPT      4              // columns per thread = K_/NTHREADS
#define NBUF     30             // LDS x-row ring slots: 30*4KB x + 2KB bnd = 124,928 B
#define ROWBYTES (K_ * 4)

// CDNA5 async copy, GV form: vdst = LDS byte-offset VGPR, vaddr = 64-bit address pair.
__device__ __forceinline__ void async_load_b128_gv(const float* gsrc, uint32_t lds_off) {
    uint64_t ga = (uint64_t)(uintptr_t)gsrc;
    asm volatile("global_load_async_to_lds_b128 %0, %1, off"
                 :: "v"(lds_off), "v"(ga)
                 : "memory");
}

// CDNA5 async copy, GVS form: uniform SGPR-pair base + 32-bit per-lane offset.
// Keeps 64-bit address math on the scalar unit in the hot loop.
__device__ __forceinline__ void async_load_b128_gvs(uint64_t gbase, uint32_t voff,
                                                    uint32_t lds_off) {
    asm volatile("global_load_async_to_lds_b128 %0, %1, %2"
                 :: "v"(lds_off), "v"(voff), "s"(gbase)
                 : "memory");
}

__device__ __forceinline__ void wait_async_le_nbufm1() {
    asm volatile("s_wait_asynccnt %0" :: "n"(NBUF - 1));
}

__global__ __launch_bounds__(NTHREADS, 1)
void maxplus_scan_gfx1250(const float* __restrict__ x, float* __restrict__ out) {
    // Dynamic LDS: [0, NBUF*K_) x-row ring, then [.., +2*NTHREADS) boundary ping-pong.
    extern __shared__ float smem[];
    float* xbuf = smem;
    float* bnd  = smem + NBUF * K_;

    const int      tid    = threadIdx.x;
    const int      b      = blockIdx.x;
    const size_t   base   = (size_t)b * T_ * K_;
    const uint32_t lds0   = __builtin_amdgcn_groupstaticsize();   // dynamic-LDS base
    const uint32_t lane16 = (uint32_t)tid * 16u;                  // this thread's 16B in a row

    // ---- prime the async pipeline: rows 0..NBUF-1 into slots 0..NBUF-1 ----
    for (int r = 0; r < NBUF; ++r) {
        async_load_b128_gv(x + base + (size_t)r * K_ + CPT * tid,
                           lds0 + (uint32_t)r * ROWBYTES + lane16);
    }

    float c0, c1, c2, c3;   // carry for columns 4*tid .. 4*tid+3

    // ---- step 0: out[0] = x[0] (no refill here; slot 0 is refilled at step 1) ----
    {
        wait_async_le_nbufm1();                    // row 0 resident (this wave's slice)
        const float4 xv = *reinterpret_cast<const float4*>(&xbuf[CPT * tid]);
        c0 = xv.x; c1 = xv.y; c2 = xv.z; c3 = xv.w;
        *reinterpret_cast<float4*>(out + base + CPT * tid) = make_float4(c0, c1, c2, c3);
        bnd[1 * NTHREADS + tid] = c3;              // old value for step 1 (parity 1)
        __syncthreads();                           // flushes DScnt: slot 0 read complete
    }

    // Uniform (scalar) pointer to the next row to prefetch; one s_add per step.
    uint64_t gnext = (uint64_t)(uintptr_t)(x + base + (size_t)NBUF * K_);

    int slot  = 1;   // ring slot holding row t
    int fslot = 0;   // ring slot freed at step t-1 -> refill target

    // ---- steps 1..T-1 ----
    #pragma unroll 1
    for (int t = 1; t < T_; ++t) {
        // Refill the slot freed last step (its ds_load completed before last barrier).
        // Issued before any waits so the fetch overlaps this whole step.
        if (t <= T_ - NBUF) {
            async_load_b128_gvs(gnext, lane16, lds0 + (uint32_t)fslot * ROWBYTES + lane16);
            gnext += ROWBYTES;
        }
        if (++fslot == NBUF) fslot = 0;

        wait_async_le_nbufm1();                    // row t resident (this wave's slice)
        const float4 xv = *reinterpret_cast<const float4*>(&xbuf[slot * K_ + CPT * tid]);

        // Neighbor's old last column (previous step), -inf at the global left edge.
        const float left = (tid == 0) ? -__builtin_inff()
                                      : bnd[(t & 1) * NTHREADS + tid - 1];
        const float n0 = xv.x + fmaxf(c0, left);
        const float n1 = xv.y + fmaxf(c1, c0);
        const float n2 = xv.z + fmaxf(c2, c1);
        const float n3 = xv.w + fmaxf(c3, c2);
        c0 = n0; c1 = n1; c2 = n2; c3 = n3;

        bnd[((t + 1) & 1) * NTHREADS + tid] = c3;  // publish for step t+1
        *reinterpret_cast<float4*>(out + base + (size_t)t * K_ + CPT * tid)
            = make_float4(c0, c1, c2, c3);         // coalesced global_store_b128

        __syncthreads();                           // orders bnd ping-pong + flushes DScnt

        if (++slot == NBUF) slot = 0;
    }
}

extern "C" void kernel_launch(void* const* d_in, const int* in_sizes, int n_in,
                              void* d_out, int out_size, void* d_ws, size_t ws_size,
                              hipStream_t stream) {
    (void)in_sizes; (void)n_in; (void)out_size; (void)d_ws; (void)ws_size;
    const float* x  = (const float*)d_in[0];
    float*      out = (float*)d_out;

    const int shbytes = (NBUF * K_ + 2 * NTHREADS) * (int)sizeof(float);  // 124,928 B

    // Opt in to >64KB dynamic LDS (gfx1250 WGP has 320KB). Idempotent, capture-safe.
    (void)hipFuncSetAttribute((const void*)maxplus_scan_gfx1250,
                              hipFuncAttributeMaxDynamicSharedMemorySize, shbytes);

    hipLaunchKernelGGL(maxplus_scan_gfx1250,
                       dim3(B_), dim3(NTHREADS), (size_t)shbytes, stream,
                       x, out);
}